// WaveletModel_79860621902296
// MI455X (gfx1250) — compile-verified
//
#include <hip/hip_runtime.h>

typedef __attribute__((ext_vector_type(2))) float v2f;
typedef __attribute__((ext_vector_type(8))) float v8f;

#define Bsz   8      // batch
#define Csz   16     // channels (== scales, diagonal pairing)
#define Lsz   1024   // H*W
#define OCT   128    // output positions handled per block
#define TPB   256    // 8 wave32

__global__ __launch_bounds__(TPB) void wavelet_wmma_kernel(
    const float* __restrict__ x,        // [8,16,1024]
    const float* __restrict__ scales,   // [16]
    const float* __restrict__ f_mod,    // [10]
    const float* __restrict__ poly_mod, // [10,3]
    const int*   __restrict__ iidx,     // [1]
    float*       __restrict__ out)      // [8,256,1024]
{
    __shared__ __align__(16) float xlds[Bsz * Lsz];   // x rows for this channel (32 KB)
    __shared__ __align__(16) float wpad[2048];        // zero-padded wavelet (8 KB)
    __shared__ __align__(16) float zlds[1024];        // zero row for M=8..15 (4 KB)
    __shared__ __align__(16) float ostage[Bsz * OCT]; // distinct results (4 KB)

    const int tid    = threadIdx.x;
    const int d      = blockIdx.x & 15;        // distinct channel (= g mod 16)
    const int o_base = (blockIdx.x >> 4) * OCT;
    const int i      = iidx[0];

    // Wavelet parameters: sigma = f_mod[i]*scale/(2*pi); u = (t/sigma)^2
    const float scale    = scales[d];
    const float fm       = f_mod[i];
    const float p0       = poly_mod[3 * i + 0];
    const float p1       = poly_mod[3 * i + 1];
    const float p2       = poly_mod[3 * i + 2];
    const float invSigma = 6.28318530717958647692f / (fm * scale);

    // wpad[t] = wav[d, t-512] for t in [512,1536), else 0.
    // Then B[j,o] = wav[d, j-o+511] = wpad[j - o + 1023].
    for (int t = tid; t < 2048; t += TPB) {
        int k = t - 512;
        float w = 0.0f;
        if ((unsigned)k < (unsigned)Lsz) {
            float r    = (float)k * invSigma;
            float u    = r * r;
            float u2   = u * u;
            float poly = 1.0f - p0 * u + p1 * u2 + p2 * u2 * u;
            w = poly * expf(-0.5f * u);
        }
        wpad[t] = w;
    }
    {   // zero row (float4) + x staging (float4, coalesced b128 loads)
        float4 zv = make_float4(0.f, 0.f, 0.f, 0.f);
        float4* z4 = (float4*)zlds;
        for (int t = tid; t < 1024 / 4; t += TPB) z4[t] = zv;

        const float4* x4 = (const float4*)(x + d * Lsz);  // row d of each batch
        float4* xl4 = (float4*)xlds;
        for (int q = tid; q < (Bsz * Lsz) / 4; q += TPB) {
            int b = q >> 8;                // 256 float4 per row
            int c = q & 255;
            xl4[q] = x4[b * (Csz * Lsz / 4) + c];
        }
    }
    __syncthreads();

    // ---- WMMA compute: M=batch(16, 8 live), N=16 output positions, K step 4 ----
    const int wave = __builtin_amdgcn_readfirstlane(tid >> 5);  // wave-uniform (SGPR)
    const int lane = tid & 31;
    const int half = lane >> 4;   // 0: lanes 0-15, 1: lanes 16-31
    const int lrow = lane & 15;
    const int o0   = o_base + wave * 16;   // uniform -> scalar loop bounds

    // A row pointer: M = lrow (batch), rows 8..15 are zero
    const float* arow = (lrow < 8) ? (xlds + (lrow << 10)) : zlds;
    const int bofs = 1023 - (o0 + lrow);   // wpad index = kk + bofs

    // Toeplitz band nonzero for k0 in [o0-514, o0+527]; align to 16 so the
    // trip count is a multiple of 4 (padding contributes exact zeros).
    int kstart = o0 - 514; if (kstart < 0) kstart = 0; kstart &= ~15;
    int kend   = (o0 + 531 + 15) & ~15;    if (kend > Lsz) kend = Lsz;

    v8f acc = {};
#pragma unroll 4
    for (int k0 = kstart; k0 < kend; k0 += 4) {
        int kk = k0 + (half << 1);          // K rows {0,1} / {2,3} per half-wave
        v2f a = *(const v2f*)(arow + kk);   // A[m=lrow, kk], A[m, kk+1]
        v2f bm;
        bm.x = wpad[kk + bofs];             // B[kk,   o0+lrow]
        bm.y = wpad[kk + bofs + 1];         // B[kk+1, o0+lrow]
        acc = __builtin_amdgcn_wmma_f32_16x16x4_f32(
            /*neg_a=*/false, a, /*neg_b=*/false, bm,
            /*c_mod=*/(short)0, acc, /*reuse_a=*/false, /*reuse_b=*/false);
    }

    // D layout: VGPR r, lanes 0-15 -> (M=r, N=lane); lanes 16-31 hold M=8..15 (zero rows)
    if (half == 0) {
#pragma unroll
        for (int r = 0; r < Bsz; ++r)
            ostage[(r << 7) + (wave << 4) + lrow] = acc[r];
    }
    __syncthreads();

    // ---- Replicate distinct result to the 16 output-channel copies (coalesced) ----
    const float4* src = (const float4*)ostage;
    for (int c2 = tid; c2 < Bsz * 16 * (OCT / 4); c2 += TPB) {
        int q   = c2 & 31;          // float4 index within OCT segment
        int rep = (c2 >> 5) & 15;   // which replica
        int b   = c2 >> 9;          // batch
        float4 v = src[(b << 5) + q];
        int g = d + (rep << 4);     // g mod 16 == d
        long off = ((long)(b * 256 + g)) * Lsz + o_base + (q << 2);
        *(float4*)(out + off) = v;
    }
}

extern "C" void kernel_launch(void* const* d_in, const int* in_sizes, int n_in,
                              void* d_out, int out_size, void* d_ws, size_t ws_size,
                              hipStream_t stream) {
    const float* x        = (const float*)d_in[0];
    const float* scales   = (const float*)d_in[1];
    const float* f_mod    = (const float*)d_in[2];
    const float* poly_mod = (const float*)d_in[3];
    const int*   iidx     = (const int*)d_in[4];
    float*       out      = (float*)d_out;

    dim3 grid(Csz * (Lsz / OCT));  // 16 channels x 8 octants = 128 blocks
    dim3 block(TPB);
    hipLaunchKernelGGL(wavelet_wmma_kernel, grid, block, 0, stream,
                       x, scales, f_mod, poly_mod, iidx, out);
}